// TransCell_76270029242484
// MI455X (gfx1250) — compile-verified
//
#include <hip/hip_runtime.h>

typedef __attribute__((ext_vector_type(2))) float v2f;
typedef __attribute__((ext_vector_type(8))) float v8f;

#define Bb 32
#define Nn 100
#define Ee 7
#define Hh 1024
#define Dd 256

// ---------------------------------------------------------------------------
// Kernel 1: qp = q @ Wq + bq   (32x1024 @ 1024x256, f32)
// via V_WMMA_F32_16X16X4_F32. Grid = 32 tiles (2 Mtiles x 16 Ntiles),
// block = 32 threads = exactly one wave (EXEC all-ones, required for WMMA).
// ---------------------------------------------------------------------------
__global__ void qp_wmma_kernel(const float* __restrict__ q,
                               const float* __restrict__ Wq,
                               const float* __restrict__ bq,
                               float* __restrict__ qp) {
  const int tile  = blockIdx.x;      // 0..31
  const int mtile = tile >> 4;       // 0..1
  const int ntile = tile & 15;       // 0..15
  const int lane  = threadIdx.x;     // 0..31
  const int ml    = lane & 15;
  const int kb    = (lane >> 4) << 1;          // K sub-offset: 0 or 2
  const int ncol  = ntile * 16 + ml;

  const float* qrow = q + (mtile * 16 + ml) * Hh;

  v8f c = {};
  for (int k = 0; k < Hh; k += 4) {
    v2f a, b;
    // A 16x4: VGPR0 = K=kb, VGPR1 = K=kb+1 (lanes 0-15: K 0/1, lanes 16-31: K 2/3)
    a.x = qrow[k + kb];
    a.y = qrow[k + kb + 1];
    // B 4x16: VGPR0 = row K=kb at column ncol, VGPR1 = row K=kb+1
    b.x = Wq[(k + kb) * Dd + ncol];
    b.y = Wq[(k + kb + 1) * Dd + ncol];
    c = __builtin_amdgcn_wmma_f32_16x16x4_f32(false, a, false, b, (short)0, c,
                                              false, false);
  }

  const int mo = (lane >> 4) << 3;   // C/D: lanes 16-31 hold rows M=r+8
#pragma unroll
  for (int r = 0; r < 8; ++r) {
    const int row = mtile * 16 + mo + r;
    qp[row * Dd + ncol] = c[r] + bq[ncol];
  }
}

// ---------------------------------------------------------------------------
// Kernel 2: one block per batch b (256 threads).
//   w[b] = Wr @ (qp[b] * Wo),  c[b] = br . (qp[b]*Wo) + bo
//   rel[i,j] = trans_mat[b,i,j,:] . w + c   (masked)
//   column softmax over i, then logits[b,i] = sum_j softmax * z[b,j]
// ---------------------------------------------------------------------------
__global__ void rel_softmax_kernel(const float* __restrict__ trans_mat,
                                   const float* __restrict__ r_mask,
                                   const float* __restrict__ z_logits,
                                   const float* __restrict__ Wr,
                                   const float* __restrict__ br,
                                   const float* __restrict__ Wo,
                                   const float* __restrict__ bo,
                                   const float* __restrict__ qp,
                                   float* __restrict__ out) {
  const int b   = blockIdx.x;
  const int tid = threadIdx.x;       // 0..255

  __shared__ float relS[Nn * 101];   // padded stride 101 -> conflict-free
  __shared__ float red[256];
  __shared__ float wS[Ee];
  __shared__ float cS;
  __shared__ float zinv[Nn];

  // v_d = qp[b][d] * Wo[d]
  const float vd = qp[b * Dd + tid] * Wo[tid];

  // 8 block reductions: w[0..6] and c
  for (int e = 0; e < 8; ++e) {
    red[tid] = (e < Ee) ? Wr[e * Dd + tid] * vd : br[tid] * vd;
    __syncthreads();
    for (int s = 128; s > 0; s >>= 1) {
      if (tid < s) red[tid] += red[tid + s];
      __syncthreads();
    }
    if (tid == 0) {
      if (e < Ee) wS[e] = red[0];
      else        cS   = red[0] + bo[0];
    }
    __syncthreads();
  }

  const float w0 = wS[0], w1 = wS[1], w2 = wS[2], w3 = wS[3];
  const float w4 = wS[4], w5 = wS[5], w6 = wS[6];
  const float cc = cS;

  // Pass 1: stream trans_mat once (HBM-bound), masked rel -> LDS
  const float* tb = trans_mat + (size_t)b * Nn * Nn * Ee;
  const float* mb = r_mask + (size_t)b * Nn * Nn;
  for (int idx = tid; idx < Nn * Nn; idx += 256) {
    const float* t = tb + idx * Ee;
    __builtin_prefetch(tb + (idx + 256) * Ee, 0, 1);  // global_prefetch_b8
    float r = cc + t[0] * w0 + t[1] * w1 + t[2] * w2 + t[3] * w3 +
              t[4] * w4 + t[5] * w5 + t[6] * w6;
    if (mb[idx] == 0.0f) r = -__builtin_inff();
    const int i = idx / Nn;
    const int j = idx - i * Nn;
    relS[i * 101 + j] = r;
  }
  __syncthreads();

  // Pass 2: per-column (axis=1) softmax stats; overwrite relS with exp terms
  if (tid < Nn) {
    const int j = tid;
    float m = -__builtin_inff();
    for (int i = 0; i < Nn; ++i) m = fmaxf(m, relS[i * 101 + j]);
    float s = 0.0f;
    for (int i = 0; i < Nn; ++i) {
      const float e = __expf(relS[i * 101 + j] - m);
      relS[i * 101 + j] = e;
      s += e;
    }
    zinv[j] = z_logits[b * Nn + j] / s;
  }
  __syncthreads();

  // Pass 3: logits[b][i] = sum_j exp[i][j] * (z[j]/S[j])
  if (tid < Nn) {
    const int i = tid;
    float acc = 0.0f;
    for (int j = 0; j < Nn; ++j) acc += relS[i * 101 + j] * zinv[j];
    out[b * Nn + i] = acc;
  }
}

// ---------------------------------------------------------------------------
extern "C" void kernel_launch(void* const* d_in, const int* in_sizes, int n_in,
                              void* d_out, int out_size, void* d_ws, size_t ws_size,
                              hipStream_t stream) {
  const float* q         = (const float*)d_in[0];
  const float* trans_mat = (const float*)d_in[1];
  const float* r_mask    = (const float*)d_in[2];
  const float* z_logits  = (const float*)d_in[3];
  const float* Wr        = (const float*)d_in[4];
  const float* br        = (const float*)d_in[5];
  const float* Wq        = (const float*)d_in[6];
  const float* bq        = (const float*)d_in[7];
  const float* Wo        = (const float*)d_in[8];
  const float* bo        = (const float*)d_in[9];

  float* qp  = (float*)d_ws;               // 32*256 floats = 32 KB scratch
  float* out = (float*)d_out;              // (32,100) f32

  qp_wmma_kernel<<<dim3(32), dim3(32), 0, stream>>>(q, Wq, bq, qp);
  rel_softmax_kernel<<<dim3(Bb), dim3(256), 0, stream>>>(
      trans_mat, r_mask, z_logits, Wr, br, Wo, bo, qp, out);
}